// HRRSelfAttention_9861244912045
// MI455X (gfx1250) — compile-verified
//
#include <hip/hip_runtime.h>
#include <math.h>

// ---------------- problem constants ----------------
#define DD   1024          // feature dim
#define SS   2048          // sequence length
#define BB   4             // batch
#define MR   (BB*SS)       // 8192 rows
#define NF   513           // rfft bins
#define NFP  528           // padded bins (multiple of 16)
#define N1   (6*NFP)       // 3168: [q_re|q_im|k_re|k_im|v_re|v_im] spectral planes
#define K2   (2*NFP)       // 1056: [G_re|G_im] for output GEMM
#define NCH  16            // scan chunks per sequence
#define CHL  (SS/NCH)      // 128 elements per chunk

typedef __attribute__((ext_vector_type(2))) float v2f;
typedef __attribute__((ext_vector_type(8))) float v8f;
typedef __attribute__((ext_vector_type(4))) int   v4i;

// ---------------- CDNA5 async global->LDS copy (probe-guarded) ----------------
#if defined(__has_builtin)
#if __has_builtin(__builtin_amdgcn_global_load_async_to_lds_b128)
#define HAVE_ASYNC_LDS 1
#endif
#endif

typedef __attribute__((address_space(1))) v4i* g4p_t;   // global int4*
typedef __attribute__((address_space(3))) v4i* l4p_t;   // LDS int4*

__device__ __forceinline__ void cp16(float* lds_dst, const float* gsrc) {
#ifdef HAVE_ASYNC_LDS
    __builtin_amdgcn_global_load_async_to_lds_b128(
        (g4p_t)(void*)(gsrc), (l4p_t)(void*)(lds_dst), 0, 0);
#else
    *(float4*)lds_dst = *(const float4*)gsrc;
#endif
}

__device__ __forceinline__ void async_wait0() {
#ifdef HAVE_ASYNC_LDS
    asm volatile("s_wait_asynccnt 0x0" ::: "memory");
#endif
}

// ---------------- workspace layout (floats) ----------------
constexpr size_t W1_OFF = 0;                         // spectral qkv weights [1024][3168]
constexpr size_t W1_SZ  = (size_t)DD * N1;           // 3,244,032
constexpr size_t W2_OFF = W1_OFF + W1_SZ;            // spectral output weights [1056][1024]
constexpr size_t W2_SZ  = (size_t)K2 * DD;           // 1,081,344
constexpr size_t F_OFF  = W2_OFF + W2_SZ;            // spectral activations [8192][3168]
constexpr size_t F_SZ   = (size_t)MR * N1;           // 25,952,256
constexpr size_t G_OFF  = F_OFF + F_SZ;              // unbound memory [8192][1056]
constexpr size_t G_SZ   = (size_t)MR * K2;           // 8,650,752
constexpr size_t CS_OFF = G_OFF + G_SZ;              // chunk sums [4][16][528][2]
constexpr size_t CS_SZ  = (size_t)BB * NCH * NFP * 2;

// ---------------- zero spectral weight buffers (padding rows/cols must be 0) --------
__global__ __launch_bounds__(256) void zero_ws_kernel(float* w1, float* w2) {
    size_t i = (size_t)blockIdx.x * 256 + threadIdx.x;
    if (i < W1_SZ) w1[i] = 0.0f;
    if (i < W2_SZ) w2[i] = 0.0f;
}

// ---------------- DFT of q/k/v weight columns -> W1 ----------------
// Fq[b,s,f] = sum_d x[b,s,d] * C[d,f],  C[d,f] = sum_e W[e,d] e^{-2pi i e f / D}
__global__ __launch_bounds__(256)
void dft_qkv_kernel(const float* __restrict__ Wq, const float* __restrict__ Wk,
                    const float* __restrict__ Wv, float* __restrict__ W1) {
    __shared__ float ct[DD], st[DD];
    for (int i = threadIdx.x; i < DD; i += 256) {
        float ang = 6.28318530717958647692f * (float)i / (float)DD;
        float s, c; sincosf(ang, &s, &c);
        ct[i] = c; st[i] = s;
    }
    __syncthreads();
    const int d = blockIdx.x * 256 + threadIdx.x;   // 0..1023
    const int f = blockIdx.y;                       // 0..512
    const int m = blockIdx.z;                       // 0=q 1=k 2=v
    const float* W = (m == 0) ? Wq : (m == 1) ? Wk : Wv;
    float sr = 0.0f, si = 0.0f;
    for (int e = 0; e < DD; ++e) {
        float w = W[(size_t)e * DD + d];            // coalesced across lanes (d)
        int idx = (e * f) & (DD - 1);
        sr = fmaf(w, ct[idx], sr);
        si = fmaf(w, st[idx], si);
    }
    W1[(size_t)d * N1 + (size_t)(2 * m) * NFP + f]     = sr;    // Re
    W1[(size_t)d * N1 + (size_t)(2 * m + 1) * NFP + f] = -si;   // Im (e^{-i})
}

// ---------------- DFT of Wo rows -> W2 = [SA ; SB] ----------------
// out[e] = sum_f Gre[f]*SA[f][e] + Gim[f]*SB[f][e]
__global__ __launch_bounds__(256)
void dft_out_kernel(const float* __restrict__ Wo, float* __restrict__ W2) {
    __shared__ float ct[DD], st[DD];
    for (int i = threadIdx.x; i < DD; i += 256) {
        float ang = 6.28318530717958647692f * (float)i / (float)DD;
        float s, c; sincosf(ang, &s, &c);
        ct[i] = c; st[i] = s;
    }
    __syncthreads();
    const int e = blockIdx.x * 256 + threadIdx.x;   // 0..1023
    const int f = blockIdx.y;                       // 0..512
    const float cf = (f == 0 || f == DD / 2) ? 1.0f : 2.0f;
    const float scale = cf / (float)DD;
    float sr = 0.0f, si = 0.0f;
    for (int d = 0; d < DD; ++d) {
        float w = Wo[(size_t)e * DD + d];
        int idx = (d * f) & (DD - 1);
        sr = fmaf(w, ct[idx], sr);
        si = fmaf(w, st[idx], si);
    }
    W2[(size_t)f * DD + e]         = scale * sr;    // SA row f
    W2[(size_t)(NFP + f) * DD + e] = -scale * si;   // SB row f
}

// ---------------- f32 WMMA GEMM: C[M,N] = A[M,K] * B[K,N], all row-major -----------
// lda == K, ldb == ldc == N. M%128==0, N%BN==0, K%16==0. 256 threads = 8 waves.
// Double-buffered LDS, async global->LDS staging overlapped with WMMA compute.
template <int BN>
__global__ __launch_bounds__(256)
void gemm_wmma_f32(const float* __restrict__ A, const float* __restrict__ Bm,
                   float* __restrict__ C, int Mdim, int Ndim, int Kdim) {
    constexpr int BM = 128;
    constexpr int NT = BN / 16;
    constexpr int LDA_S = 20;          // 80B rows: b128-aligned, conflict-free frag reads
    __shared__ float sA[2][BM * LDA_S];
    __shared__ float sB[2][16 * BN];

    const int tid  = threadIdx.x;
    const int wave = tid >> 5;                      // 0..7 -> M-tile
    const int lane = tid & 31;
    const int half = lane >> 4;
    const int l16  = lane & 15;

    const int bn0 = blockIdx.x * BN;
    const int bm0 = blockIdx.y * BM;

    // stage one 16-wide K chunk into buffer `buf` (all addresses 16B aligned)
    auto issue = [&](int buf, int kc) {
        // A tile: 128x16 floats = 512 x b128 -> 2 per thread
        const int r  = tid >> 2;
        const int c4 = (tid & 3) * 4;
        cp16(&sA[buf][r * LDA_S + c4],
             A + (size_t)(bm0 + r) * Kdim + kc + c4);
        cp16(&sA[buf][(r + 64) * LDA_S + c4],
             A + (size_t)(bm0 + r + 64) * Kdim + kc + c4);
        // B tile: 16xBN floats = (16*BN/4) x b128
        constexpr int NB = (16 * BN) / 4;           // 192 or 256
        if (tid < NB) {
            const int rb  = tid / (BN / 4);
            const int cb4 = (tid % (BN / 4)) * 4;
            cp16(&sB[buf][rb * BN + cb4],
                 Bm + (size_t)(kc + rb) * Ndim + bn0 + cb4);
        }
    };

    v8f acc[NT] = {};
    const int NK = Kdim >> 4;

    issue(0, 0);
    async_wait0();
    __syncthreads();

    for (int kci = 0; kci < NK; ++kci) {
        const int cur = kci & 1;
        if (kci + 1 < NK) issue(cur ^ 1, (kci + 1) << 4);

        const float* a_s = sA[cur];
        const float* b_s = sB[cur];
#pragma unroll
        for (int kk = 0; kk < 4; ++kk) {
            const int ak = 4 * kk + 2 * half;       // K pair this half-wave owns
            v2f a;
            a.x = a_s[(16 * wave + l16) * LDA_S + ak];
            a.y = a_s[(16 * wave + l16) * LDA_S + ak + 1];
#pragma unroll
            for (int t = 0; t < NT; ++t) {
                v2f b;
                b.x = b_s[ak * BN + 16 * t + l16];
                b.y = b_s[(ak + 1) * BN + 16 * t + l16];
                acc[t] = __builtin_amdgcn_wmma_f32_16x16x4_f32(
                    false, a, false, b, (short)0, acc[t], false, false);
            }
        }
        async_wait0();      // own async stores into nxt buffer landed
        __syncthreads();    // all waves done reading cur + publishing nxt
    }

    // store per documented 16x16 f32 C/D layout
#pragma unroll
    for (int t = 0; t < NT; ++t) {
        const int col = bn0 + 16 * t + l16;
#pragma unroll
        for (int i = 0; i < 8; ++i) {
            const int row = bm0 + 16 * wave + i + 8 * half;
            C[(size_t)row * Ndim + col] = acc[t][i];
        }
    }
}

// ---------------- causal scan of Fk*Fv, 3-pass chunked ----------------
__global__ __launch_bounds__(256)
void bind_pass1(const float* __restrict__ F, float* __restrict__ cs) {
    int tid = blockIdx.x * 256 + threadIdx.x;
    if (tid >= BB * NCH * NFP) return;
    int f  = tid % NFP;
    int ch = (tid / NFP) % NCH;
    int b  = tid / (NFP * NCH);
    size_t r0 = (size_t)b * SS + (size_t)ch * CHL;
    float sr = 0.0f, si = 0.0f;
    for (int s = 0; s < CHL; ++s) {
        const float* row = F + (r0 + s) * N1;
        __builtin_prefetch(&row[2 * NFP + f] + 4 * N1, 0, 0);
        __builtin_prefetch(&row[4 * NFP + f] + 4 * N1, 0, 0);
        float kr = row[2 * NFP + f], ki = row[3 * NFP + f];
        float vr = row[4 * NFP + f], vi = row[5 * NFP + f];
        sr += kr * vr - ki * vi;
        si += kr * vi + ki * vr;
    }
    cs[(size_t)tid * 2 + 0] = sr;
    cs[(size_t)tid * 2 + 1] = si;
}

__global__ __launch_bounds__(256)
void bind_pass2(float* __restrict__ cs) {
    int tid = blockIdx.x * 256 + threadIdx.x;
    if (tid >= BB * NFP) return;
    int f = tid % NFP;
    int b = tid / NFP;
    float ar = 0.0f, ai = 0.0f;
    for (int ch = 0; ch < NCH; ++ch) {
        size_t idx = ((size_t)(b * NCH + ch) * NFP + f) * 2;
        float tr = cs[idx], ti = cs[idx + 1];
        cs[idx] = ar; cs[idx + 1] = ai;             // exclusive prefix
        ar += tr; ai += ti;
    }
}

__global__ __launch_bounds__(256)
void bind_pass3(const float* __restrict__ F, const float* __restrict__ cs,
                float* __restrict__ G) {
    int tid = blockIdx.x * 256 + threadIdx.x;
    if (tid >= BB * NCH * NFP) return;
    int f  = tid % NFP;
    int ch = (tid / NFP) % NCH;
    int b  = tid / (NFP * NCH);
    size_t r0 = (size_t)b * SS + (size_t)ch * CHL;
    float sr = cs[(size_t)tid * 2 + 0];
    float si = cs[(size_t)tid * 2 + 1];
    for (int s = 0; s < CHL; ++s) {
        const float* row = F + (r0 + s) * N1;
        __builtin_prefetch(&row[f] + 4 * N1, 0, 0);
        __builtin_prefetch(&row[2 * NFP + f] + 4 * N1, 0, 0);
        __builtin_prefetch(&row[4 * NFP + f] + 4 * N1, 0, 0);
        float kr = row[2 * NFP + f], ki = row[3 * NFP + f];
        float vr = row[4 * NFP + f], vi = row[5 * NFP + f];
        sr += kr * vr - ki * vi;                    // running Fmem
        si += kr * vi + ki * vr;
        float qr = row[f], qi = row[NFP + f];       // Fq
        G[(r0 + s) * K2 + f]       = sr * qr + si * qi;   // Fmem * conj(Fq)
        G[(r0 + s) * K2 + NFP + f] = si * qr - sr * qi;
    }
}

// ---------------- launch ----------------
extern "C" void kernel_launch(void* const* d_in, const int* in_sizes, int n_in,
                              void* d_out, int out_size, void* d_ws, size_t ws_size,
                              hipStream_t stream) {
    (void)in_sizes; (void)n_in; (void)out_size; (void)ws_size;
    const float* x  = (const float*)d_in[0];
    const float* Wq = (const float*)d_in[1];
    const float* Wk = (const float*)d_in[2];
    const float* Wv = (const float*)d_in[3];
    const float* Wo = (const float*)d_in[4];
    float* out = (float*)d_out;
    float* ws  = (float*)d_ws;

    float* W1 = ws + W1_OFF;
    float* W2 = ws + W2_OFF;
    float* F  = ws + F_OFF;
    float* G  = ws + G_OFF;
    float* CS = ws + CS_OFF;

    // 1. zero spectral-weight buffers (padding correctness)
    zero_ws_kernel<<<(int)((W1_SZ + 255) / 256), 256, 0, stream>>>(W1, W2);

    // 2. per-call weight DFTs
    dft_qkv_kernel<<<dim3(DD / 256, NF, 3), 256, 0, stream>>>(Wq, Wk, Wv, W1);
    dft_out_kernel<<<dim3(DD / 256, NF), 256, 0, stream>>>(Wo, W2);

    // 3. spectral projection: F = x @ W1   [8192x1024]x[1024x3168]
    gemm_wmma_f32<48><<<dim3(N1 / 48, MR / 128), 256, 0, stream>>>(x, W1, F, MR, N1, DD);

    // 4. bind + causal cumulative memory + unbind -> G
    bind_pass1<<<(BB * NCH * NFP) / 256, 256, 0, stream>>>(F, CS);
    bind_pass2<<<(BB * NFP + 255) / 256, 256, 0, stream>>>(CS);
    bind_pass3<<<(BB * NCH * NFP) / 256, 256, 0, stream>>>(F, CS, G);

    // 5. spectral output projection: out = G @ W2   [8192x1056]x[1056x1024]
    gemm_wmma_f32<64><<<dim3(DD / 64, MR / 128), 256, 0, stream>>>(G, W2, out, MR, DD, K2);
}